// NormalizingFlowModel_3676492006195
// MI455X (gfx1250) — compile-verified
//
#include <hip/hip_runtime.h>

// ---------------------------------------------------------------------------
// RealNVP normalizing flow, CDNA5 (gfx1250) WMMA implementation, round 3.
//
// Per 16-sample tile each layer is a chain of f32 WMMAs:
//   D(16x16) = A(16x4: W^T fragment) x B(4x16: X^T fragment) + C
// Two coupling MLPs fused into 16 output columns (block-diagonal hidden).
// 4 tiles (64 samples) per wave -> 4 independent WMMA accumulation chains and
// 4x amortization of weight-fragment gathers.
//
// pi-folding: feeding D registers (d[2c], d[2c+1]) directly as B chunk c
// presents Y rows {2c,2c+1,2c+8,2c+9} at K positions {4c..4c+3}; the per-lane
// A gather compensates by fetching weight ROW (2c + v + (hiHalf?8:0)).
// No cross-lane shuffles between chained matmuls.
// ---------------------------------------------------------------------------

#define BTOT   262144
#define DD     16
#define LLAY   15
#define NHID   3
#define EPSV   1e-6f
#define SLOPEV 0.01f

#define WPB    4          // waves per block
#define TPW    4          // 16-sample tiles per wave
#define THREADS (WPB * 32)
#define SAMP_PER_BLOCK (WPB * TPW * 16)   // 256

typedef __attribute__((ext_vector_type(2))) float v2f;
typedef __attribute__((ext_vector_type(8))) float v8f;

__device__ __forceinline__ v8f wmma4(float a0, float a1, float b0, float b1, v8f c) {
    v2f A; A.x = a0; A.y = a1;
    v2f B; B.x = b0; B.y = b1;
    return __builtin_amdgcn_wmma_f32_16x16x4_f32(false, A, false, B, (short)0, c,
                                                 false, false);
}

// leaky_relu(x) = max(x, slope*x), raw v_max_num avoids fmaxf canonicalization
__device__ __forceinline__ float leaky(float x) {
    float s = x * SLOPEV;
    float y;
    asm("v_max_num_f32 %0, %1, %2" : "=v"(y) : "v"(x), "v"(s));
    return y;
}

// LDS weight image: rows padded to stride 9 floats (bank-conflict free gathers)
#define O_WIN  0        // 15*2   matrices * 72 = 2160
#define O_WHID 2160     // 15*2*3 matrices * 72 = 6480
#define O_WOUT 8640     // 15*2   matrices * 72 = 2160
#define N_SW   10800
#define STG    17                     // staging row stride (padded)
#define TILE_STG (16 * STG)           // 272 floats per 16x16 tile

__global__ __launch_bounds__(THREADS)
void flow_wmma_kernel(const float* __restrict__ z,
                      const float* __restrict__ W_in,  const float* __restrict__ b_in,
                      const float* __restrict__ W_hid, const float* __restrict__ b_hid,
                      const float* __restrict__ W_out, const float* __restrict__ b_out,
                      const int*   __restrict__ perms,
                      float* __restrict__ y_out, float* __restrict__ ld_out)
{
    __shared__ float sW[N_SW];
    __shared__ float sStage[WPB * TPW * TILE_STG];   // 4*4*272 = 4352 floats

    // Cooperative weight staging, row-pad transform (mat,k,h) -> mat*72+k*9+h
    for (int i = threadIdx.x; i < 1920; i += THREADS)
        sW[O_WIN  + (i >> 3) * 9 + (i & 7)] = W_in[i];
    for (int i = threadIdx.x; i < 5760; i += THREADS)
        sW[O_WHID + (i >> 3) * 9 + (i & 7)] = W_hid[i];
    for (int i = threadIdx.x; i < 1920; i += THREADS)
        sW[O_WOUT + (i >> 3) * 9 + (i & 7)] = W_out[i];
    __syncthreads();

    const int  lane  = threadIdx.x & 31;
    const int  wave  = threadIdx.x >> 5;
    const int  n     = lane & 15;           // sample column / output column
    const bool hiH   = lane >= 16;
    const int  hioff = hiH ? 8 : 0;
    const int  sMlp  = (lane >> 3) & 1;     // which MLP this output column is
    const int  hcol  = lane & 7;            // unit within that MLP
    const int  base  = (blockIdx.x * WPB + wave) * (TPW * 16);  // 64 samples/wave

    // z^T in C/D layout: zreg[t][r] = Z[r + hioff][n]
    float zreg[TPW][8];
    float ldacc[TPW];
#pragma unroll
    for (int t = 0; t < TPW; ++t) {
        ldacc[t] = 0.f;
#pragma unroll
        for (int r = 0; r < 8; ++r)
            zreg[t][r] = z[(size_t)(base + t * 16 + n) * DD + r + hioff];
    }

#pragma unroll 1
    for (int l = 0; l < LLAY; ++l) {
        // ---------------- input matmul: C = fused b_in, B = zreg (pi-folded)
        v8f acc[TPW];
#pragma unroll
        for (int r = 0; r < 8; ++r) {
            float v = hiH ? b_in[l * 16 + 8 + r] : b_in[l * 16 + r];
#pragma unroll
            for (int t = 0; t < TPW; ++t) acc[t][r] = v;
        }
#pragma unroll
        for (int c = 0; c < 4; ++c) {
            int   r0 = 2 * c + hioff;                  // pi-folded rows r0, r0+1
            float w0 = sW[O_WIN + (l * 2 + sMlp) * 72 + (r0 & 7) * 9 + hcol];
            float w1 = sW[O_WIN + (l * 2 + sMlp) * 72 + ((r0 + 1) & 7) * 9 + hcol];
            float a0 = hiH ? 0.f : w0;                 // rows >=8 are z_r: mask
            float a1 = hiH ? 0.f : w1;
#pragma unroll
            for (int t = 0; t < TPW; ++t)
                acc[t] = wmma4(a0, a1, zreg[t][2 * c], zreg[t][2 * c + 1], acc[t]);
        }

        // ---------------- 3 hidden (block-diag) + output matmul, pi-folded
#pragma unroll
        for (int j = 0; j <= NHID; ++j) {
            const bool is_out = (j == NHID);

#pragma unroll
            for (int t = 0; t < TPW; ++t)
#pragma unroll
                for (int r = 0; r < 8; ++r)
                    acc[t][r] = leaky(acc[t][r]);

            v8f nacc[TPW];
#pragma unroll
            for (int r = 0; r < 8; ++r) {
                float blo = is_out ? b_out[l * 16 + r]
                                   : b_hid[(l * 2 + 0) * 24 + j * 8 + r];
                float bhi = is_out ? b_out[l * 16 + 8 + r]
                                   : b_hid[(l * 2 + 1) * 24 + j * 8 + r];
                float v = hiH ? bhi : blo;
#pragma unroll
                for (int t = 0; t < TPW; ++t) nacc[t][r] = v;
            }

#pragma unroll
            for (int c = 0; c < 4; ++c) {
                int  r0    = 2 * c + hioff;            // rows r0, r0+1 (same octet)
                int  baseW = is_out ? (O_WOUT + (l * 2 + sMlp) * 72)
                                    : (O_WHID + ((l * 2 + sMlp) * 3 + j) * 72);
                float w0 = sW[baseW + (r0 & 7) * 9 + hcol];
                float w1 = sW[baseW + ((r0 + 1) & 7) * 9 + hcol];
                bool  blk = ((r0 >> 3) == sMlp);       // block-diagonal mask
                float a0 = blk ? w0 : 0.f;
                float a1 = blk ? w1 : 0.f;
#pragma unroll
                for (int t = 0; t < TPW; ++t)
                    nacc[t] = wmma4(a0, a1, acc[t][2 * c], acc[t][2 * c + 1], nacc[t]);
            }
#pragma unroll
            for (int t = 0; t < TPW; ++t) acc[t] = nacc[t];
        }

        // ---------------- affine + logdet in the HIGH half (b, z_r in-lane)
        // acc low lanes = log_s[r][n], acc high lanes = b[r][n]
        if (l < LLAY - 1) {
            __syncthreads();
#pragma unroll
            for (int t = 0; t < TPW; ++t) {
                const int tb = (wave * TPW + t) * TILE_STG;
#pragma unroll
                for (int r = 0; r < 8; ++r) {
                    float d   = acc[t][r];
                    float lsw = __shfl_xor(d, 16, 32);              // high: log_s
                    float ls  = __builtin_amdgcn_fmed3f(lsw, -5.f, 5.f);
                    float yv  = (__expf(ls) + EPSV) * zreg[t][r] + d;   // high: y_r
                    ldacc[t] += ls;                                 // valid high
                    float nz  = hiH ? yv : zreg[t][r];              // [z_l ; y_r]
                    sStage[tb + (r + hioff) * STG + n] = nz;
                }
            }
            __syncthreads();
            // permutation gather for next layer
#pragma unroll
            for (int t = 0; t < TPW; ++t) {
                const int tb = (wave * TPW + t) * TILE_STG;
#pragma unroll
                for (int r = 0; r < 8; ++r) {
                    int pf = perms[l * 16 + r + hioff];
                    zreg[t][r] = sStage[tb + pf * STG + n];
                }
            }
        } else {
            // last layer: perms[L-1] == arange, combine straight into registers
#pragma unroll
            for (int t = 0; t < TPW; ++t)
#pragma unroll
                for (int r = 0; r < 8; ++r) {
                    float d   = acc[t][r];
                    float lsw = __shfl_xor(d, 16, 32);
                    float ls  = __builtin_amdgcn_fmed3f(lsw, -5.f, 5.f);
                    float yv  = (__expf(ls) + EPSV) * zreg[t][r] + d;
                    ldacc[t] += ls;
                    zreg[t][r] = hiH ? yv : zreg[t][r];
                }
        }
    }

    // ---------------- store y (B x 16) and ld (B; valid in high lanes)
#pragma unroll
    for (int t = 0; t < TPW; ++t) {
#pragma unroll
        for (int r = 0; r < 8; ++r)
            y_out[(size_t)(base + t * 16 + n) * DD + r + hioff] = zreg[t][r];
        if (hiH)
            ld_out[base + t * 16 + n] = ldacc[t];
    }
}

extern "C" void kernel_launch(void* const* d_in, const int* in_sizes, int n_in,
                              void* d_out, int out_size, void* d_ws, size_t ws_size,
                              hipStream_t stream) {
    const float* z     = (const float*)d_in[0];
    const float* W_in  = (const float*)d_in[1];
    const float* b_in  = (const float*)d_in[2];
    const float* W_hid = (const float*)d_in[3];
    const float* b_hid = (const float*)d_in[4];
    const float* W_out = (const float*)d_in[5];
    const float* b_out = (const float*)d_in[6];
    const int*   perms = (const int*)d_in[7];

    float* y  = (float*)d_out;                 // B*16 floats
    float* ld = y + (size_t)BTOT * DD;         // then B floats

    dim3 grid(BTOT / SAMP_PER_BLOCK);          // 1024 blocks
    dim3 block(THREADS);                       // 4 wave32 per block
    flow_wmma_kernel<<<grid, block, 0, stream>>>(z, W_in, b_in, W_hid, b_hid,
                                                 W_out, b_out, perms, y, ld);
}